// DyRep_49100066127993
// MI455X (gfx1250) — compile-verified
//
#include <hip/hip_runtime.h>
#include <hip/hip_bf16.h>
#include <math.h>

#define NN 8192
#define HH 64
#define BB 128
#define SMC 20
#define LIST_MAX 512
#define EPSF 1e-7f

typedef __attribute__((ext_vector_type(2))) float v2f;
typedef __attribute__((ext_vector_type(8))) float v8f;

// ---------------------------------------------------------------------------
// h_all = emb @ W_h^T + b_h   (8192x64 @ 64x64) via V_WMMA_F32_16X16X4_F32
// One wave per 16x16 output tile. A = emb tile (16x4), B = W_h^T tile (4x16).
// f32 A 16x4 layout: lane l holds row (l&15); VGPR0/1 = K = 2*(l>>4), 2*(l>>4)+1
// f32 B 4x16 layout: lane l holds col (l&15); same K split.
// D: VGPR r, lane l -> row r + 8*(l>>4), col (l&15).
// ---------------------------------------------------------------------------
__global__ __launch_bounds__(32) void hinit_wmma_kernel(
    const float* __restrict__ emb, const float* __restrict__ Wh,
    const float* __restrict__ bh, float* __restrict__ h_all) {
  const int tileM = blockIdx.x * 16;   // over N
  const int tileN = blockIdx.y * 16;   // over H
  const int lane  = threadIdx.x;
  const int rc    = lane & 15;
  const int kh    = lane >> 4;         // 0/1 -> which K half-pair

  const float* aRow = emb + (size_t)(tileM + rc) * HH;  // A row per lane
  const float* bRow = Wh  + (size_t)(tileN + rc) * HH;  // B[k][n] = Wh[n][k]

  v8f acc = {};
  for (int k0 = 0; k0 < HH; k0 += 4) {
    const int k = k0 + 2 * kh;
    v2f av; av[0] = aRow[k]; av[1] = aRow[k + 1];
    v2f bv; bv[0] = bRow[k]; bv[1] = bRow[k + 1];
    acc = __builtin_amdgcn_wmma_f32_16x16x4_f32(
        /*neg_a=*/false, av, /*neg_b=*/false, bv,
        /*c_mod=*/(short)0, acc, /*reuse_a=*/false, /*reuse_b=*/false);
  }
  const int col = tileN + rc;
  const float bias = bh[col];
#pragma unroll
  for (int r = 0; r < 8; ++r) {
    const int row = tileM + r + 8 * kh;
    h_all[(size_t)row * HH + col] = acc[r] + bias;
  }
}

// ---------------------------------------------------------------------------
// Intensity helper: g = 0.5*Σ (zu+zv)*(w_lo+w_hi) + b ; lam = p*log1p(exp(clip(g/p)))
// ---------------------------------------------------------------------------
__device__ __forceinline__ float intensity(const float* __restrict__ zu,
                                           const float* __restrict__ zv,
                                           const float* __restrict__ w,
                                           float b, float p) {
  float s = 0.f;
#pragma unroll 8
  for (int h = 0; h < HH; ++h) s += (zu[h] + zv[h]) * (w[h] + w[HH + h]);
  const float g = 0.5f * s + b;
  float r = g / p;
  r = fminf(fmaxf(r, -75.f), 75.f);
  return p * log1pf(expf(r));
}

__global__ __launch_bounds__(BB) void lam_kernel(
    const int* __restrict__ u, const int* __restrict__ v, const int* __restrict__ k,
    const float* __restrict__ z0, const float* __restrict__ omega_w,
    const float* __restrict__ omega_b, const float* __restrict__ psi,
    float* __restrict__ lam) {
  const int i  = threadIdx.x;
  const int ki = k[i];
  lam[i] = intensity(z0 + (size_t)u[i] * HH, z0 + (size_t)v[i] * HH,
                     omega_w + (size_t)ki * 2 * HH, omega_b[ki], psi[ki]);
}

// L_surv[s] = Σ_i (1/SMC) * Σ_kk [ inten(u_i, vo_is, kk) + inten(v_i, uo_is, kk) ]
__global__ __launch_bounds__(BB) void surv_kernel(
    const int* __restrict__ u, const int* __restrict__ v,
    const int* __restrict__ u_others, const int* __restrict__ v_others,
    const float* __restrict__ z0, const float* __restrict__ omega_w,
    const float* __restrict__ omega_b, const float* __restrict__ psi,
    float* __restrict__ Lsurv) {
  __shared__ float red[BB];
  const int s = blockIdx.x;   // 0..SMC-1
  const int i = threadIdx.x;  // 0..BB-1
  const float* zu  = z0 + (size_t)u[i] * HH;
  const float* zv  = z0 + (size_t)v[i] * HH;
  const float* zvo = z0 + (size_t)v_others[i * SMC + s] * HH;
  const float* zuo = z0 + (size_t)u_others[i * SMC + s] * HH;
  float acc = 0.f;
#pragma unroll
  for (int kk = 0; kk < 2; ++kk) {
    const float* w = omega_w + (size_t)kk * 2 * HH;
    const float b = omega_b[kk], p = psi[kk];
    acc += intensity(zu, zvo, w, b, p);
    acc += intensity(zv, zuo, w, b, p);
  }
  red[i] = acc * (1.0f / (float)SMC);
  __syncthreads();
  for (int off = BB / 2; off > 0; off >>= 1) {
    if (i < off) red[i] += red[i + off];
    __syncthreads();
  }
  if (i == 0) Lsurv[s] = red[0];
}

// ---------------------------------------------------------------------------
// init: z = embeddings (in d_out), tbar = last_event_time
// ---------------------------------------------------------------------------
__global__ __launch_bounds__(256) void init_kernel(
    const float* __restrict__ emb, const float* __restrict__ let,
    float* __restrict__ z, float* __restrict__ tbar) {
  const int i = blockIdx.x * 256 + threadIdx.x;
  if (i < NN * HH) z[i] = emb[i];
  if (i < NN) tbar[i] = let[i];
}

// ---------------------------------------------------------------------------
// Precompute neighbor lists for the 256 (event,side) row scans.
// block b: e = b>>1; n2 = (b&1) ? u[e] : v[e]   (even = v-side = pass 0)
// Emits compact (index, q/(Σq+EPS)) list + count. qsum is a fixed-order tree
// reduction (deterministic); list order only feeds an exact-commutative max.
// ---------------------------------------------------------------------------
__global__ __launch_bounds__(256) void neigh_kernel(
    const int* __restrict__ u, const int* __restrict__ v,
    const float* __restrict__ A, const float* __restrict__ S,
    int* __restrict__ lidx, float* __restrict__ lw, int* __restrict__ lcnt) {
  __shared__ float red[256];
  __shared__ int   s_cnt;
  __shared__ int   s_idx[LIST_MAX];
  __shared__ float s_q[LIST_MAX];
  const int b   = blockIdx.x;
  const int e   = b >> 1;
  const int n2  = (b & 1) ? u[e] : v[e];
  const int tid = threadIdx.x;
  if (tid == 0) s_cnt = 0;
  __syncthreads();
  const float* Arow = A + (size_t)n2 * NN;
  const float* Srow = S + (size_t)n2 * NN;
  float qs = 0.f;
  for (int c = tid; c < NN; c += 256) {
    if (Arow[c] > 0.f) {
      const float qv = expf(Srow[c]);
      qs += qv;
      const int pos = atomicAdd(&s_cnt, 1);
      if (pos < LIST_MAX) { s_idx[pos] = c; s_q[pos] = qv; }
    }
  }
  red[tid] = qs;
  __syncthreads();
  for (int off = 128; off > 0; off >>= 1) {
    if (tid < off) red[tid] += red[tid + off];
    __syncthreads();
  }
  const float inv = 1.0f / (red[0] + EPSF);
  const int cnt = (s_cnt < LIST_MAX) ? s_cnt : LIST_MAX;
  for (int j = tid; j < cnt; j += 256) {
    lidx[(size_t)b * LIST_MAX + j] = s_idx[j];
    lw[(size_t)b * LIST_MAX + j]   = s_q[j] * inv;
  }
  if (tid == 0) lcnt[b] = cnt;
}

// ---------------------------------------------------------------------------
// Sequential scan over 128 events: one 256-thread workgroup (8 wave32s).
// z lives in d_out (updated in place -> z_final). Weights cached in LDS.
// ---------------------------------------------------------------------------
__global__ __launch_bounds__(256) void scan_kernel(
    const int* __restrict__ u, const int* __restrict__ v, const float* __restrict__ t,
    const float* __restrict__ Wh, const float* __restrict__ bh,
    const float* __restrict__ WS, const float* __restrict__ WR,
    const float* __restrict__ Wt,
    const int* __restrict__ lidx, const float* __restrict__ lw,
    const int* __restrict__ lcnt,
    float* __restrict__ z, float* __restrict__ h_all, float* __restrict__ tbar) {
  __shared__ float sWS[HH * HH], sWR[HH * HH], sWh[HH * HH];
  __shared__ float sWt[HH], sbh[HH];
  __shared__ float s_h0[HH], s_h1[HH], s_newv[HH], s_newu[HH];
  __shared__ float s_pa[256], s_pb[256], s_maxp[256];

  const int tid = threadIdx.x;
  for (int i = tid; i < HH * HH; i += 256) { sWS[i] = WS[i]; sWR[i] = WR[i]; sWh[i] = Wh[i]; }
  if (tid < HH) { sWt[tid] = Wt[tid]; sbh[tid] = bh[tid]; }
  __syncthreads();

  const int h   = tid & 63;
  const int grp = tid >> 6;  // 0..3

  for (int e = 0; e < BB; ++e) {
    const int ui = u[e], vi = v[e];
    const float ti  = t[e];
    const float dt0 = ti - tbar[ui];  // used by new_v (matches reference)
    const float dt1 = ti - tbar[vi];  // used by new_u

    // agg(vi) -> s_h0 (pass 0), agg(ui) -> s_h1 (pass 1)
    for (int pass = 0; pass < 2; ++pass) {
      const int b   = 2 * e + pass;
      const int cnt = lcnt[b];
      const int*   li  = lidx + (size_t)b * LIST_MAX;
      const float* lww = lw   + (size_t)b * LIST_MAX;
      float m = 0.0f;  // identity: sigmoid > 0; cnt==0 -> result 0 (mask.any() case)
      for (int ee = grp; ee < cnt; ee += 4) {
        const float x = lww[ee] * h_all[(size_t)li[ee] * HH + h];
        m = fmaxf(m, 1.0f / (1.0f + expf(-x)));
      }
      s_maxp[tid] = m;
      __syncthreads();
      if (tid < HH) {
        float* outv = pass ? s_h1 : s_h0;
        outv[tid] = fmaxf(fmaxf(s_maxp[tid], s_maxp[64 + tid]),
                          fmaxf(s_maxp[128 + tid], s_maxp[192 + tid]));
      }
      __syncthreads();
    }

    // new_v = sig(h0@WS^T + z[vi]@WR^T + dt0*Wt); new_u likewise (h1, z[ui], dt1)
    {
      const float* zrv = z + (size_t)vi * HH;
      const float* zru = z + (size_t)ui * HH;
      float pv = 0.f, pu = 0.f;
#pragma unroll 4
      for (int jj = 0; jj < 16; ++jj) {
        const int j = grp * 16 + jj;
        const float ws = sWS[h * HH + j], wr = sWR[h * HH + j];
        pv += s_h0[j] * ws + zrv[j] * wr;
        pu += s_h1[j] * ws + zru[j] * wr;
      }
      s_pa[tid] = pv; s_pb[tid] = pu;
      __syncthreads();
      if (tid < HH) {
        float sv = (s_pa[tid] + s_pa[64 + tid]) + (s_pa[128 + tid] + s_pa[192 + tid]);
        float su = (s_pb[tid] + s_pb[64 + tid]) + (s_pb[128 + tid] + s_pb[192 + tid]);
        sv += dt0 * sWt[tid];
        su += dt1 * sWt[tid];
        s_newv[tid] = 1.0f / (1.0f + expf(-sv));
        s_newu[tid] = 1.0f / (1.0f + expf(-su));
      }
      __syncthreads();
    }

    // h-rows: hv = newv@Wh^T + bh ; hu = newu@Wh^T + bh ; commit state
    {
      float pv = 0.f, pu = 0.f;
#pragma unroll 4
      for (int jj = 0; jj < 16; ++jj) {
        const int j = grp * 16 + jj;
        const float wh = sWh[h * HH + j];
        pv += s_newv[j] * wh;
        pu += s_newu[j] * wh;
      }
      s_pa[tid] = pv; s_pb[tid] = pu;
      __syncthreads();
      if (tid < HH) {
        const float hv = (s_pa[tid] + s_pa[64 + tid]) + (s_pa[128 + tid] + s_pa[192 + tid]) + sbh[tid];
        const float hu = (s_pb[tid] + s_pb[64 + tid]) + (s_pb[128 + tid] + s_pb[192 + tid]) + sbh[tid];
        // write vi first, then ui: same thread -> ui wins on tie (matches .at[vi].set(...).at[ui].set(...))
        z[(size_t)vi * HH + tid] = s_newv[tid];
        z[(size_t)ui * HH + tid] = s_newu[tid];
        h_all[(size_t)vi * HH + tid] = hv;
        h_all[(size_t)ui * HH + tid] = hu;
      }
      if (tid == 64) { tbar[ui] = ti; tbar[vi] = ti; }
      __syncthreads();
    }
  }
}

// ---------------------------------------------------------------------------
extern "C" void kernel_launch(void* const* d_in, const int* in_sizes, int n_in,
                              void* d_out, int out_size, void* d_ws, size_t ws_size,
                              hipStream_t stream) {
  const int*   u        = (const int*)d_in[0];
  const int*   v        = (const int*)d_in[1];
  const float* t        = (const float*)d_in[2];
  const int*   k        = (const int*)d_in[3];
  const int*   u_others = (const int*)d_in[4];
  const int*   v_others = (const int*)d_in[5];
  const float* emb      = (const float*)d_in[6];
  const float* A        = (const float*)d_in[7];
  const float* S        = (const float*)d_in[8];
  const float* let      = (const float*)d_in[9];
  const float* Wh       = (const float*)d_in[10];
  const float* bh       = (const float*)d_in[11];
  const float* WS       = (const float*)d_in[12];
  const float* WR       = (const float*)d_in[13];
  const float* Wt       = (const float*)d_in[14];
  const float* psi      = (const float*)d_in[15];
  const float* omega_w  = (const float*)d_in[16];
  const float* omega_b  = (const float*)d_in[17];

  float* lam   = (float*)d_out;          // [128]
  float* Lsurv = lam + BB;               // [20]
  float* z     = Lsurv + SMC;            // [8192*64], becomes z_final in place

  // workspace layout
  float* h_all = (float*)d_ws;                        // NN*HH floats
  float* tbar  = h_all + (size_t)NN * HH;             // NN floats
  int*   lidx  = (int*)(tbar + NN);                   // 256*LIST_MAX ints
  float* lw    = (float*)(lidx + 256 * LIST_MAX);     // 256*LIST_MAX floats
  int*   lcnt  = (int*)(lw + 256 * LIST_MAX);         // 256 ints

  lam_kernel<<<1, BB, 0, stream>>>(u, v, k, emb, omega_w, omega_b, psi, lam);
  surv_kernel<<<SMC, BB, 0, stream>>>(u, v, u_others, v_others, emb,
                                      omega_w, omega_b, psi, Lsurv);
  init_kernel<<<(NN * HH + 255) / 256, 256, 0, stream>>>(emb, let, z, tbar);
  hinit_wmma_kernel<<<dim3(NN / 16, HH / 16), 32, 0, stream>>>(emb, Wh, bh, h_all);
  neigh_kernel<<<2 * BB, 256, 0, stream>>>(u, v, A, S, lidx, lw, lcnt);
  scan_kernel<<<1, 256, 0, stream>>>(u, v, t, Wh, bh, WS, WR, Wt,
                                     lidx, lw, lcnt, z, h_all, tbar);
}